// MambaMEMScorer_83932250898894
// MI455X (gfx1250) — compile-verified
//
#include <hip/hip_runtime.h>
#include <hip/hip_bf16.h>

// ---------------- problem constants (match reference) ----------------
#define B_   16
#define L_   4096
#define F_   32
#define DM_  128
#define DS_  16
#define NL_  2
#define E_   256      // 2*DM
#define K_   4
#define R_   8
#define XPN_ 40       // R_ + 2*DS
#define BL_  (B_*L_)  // 65536 rows

typedef float v2f __attribute__((ext_vector_type(2)));
typedef float v4f __attribute__((ext_vector_type(4)));
typedef float v8f __attribute__((ext_vector_type(8)));

#define GF_BIAS 1
#define GF_ACC  2

// =====================================================================
// fp32 WMMA GEMM:  C[M,N] = A[M,K] @ W[K,N] (+bias) (+C)
//
// Block = 256 threads (8 waves), block tile = 128 rows x 64 cols.
// W K-slabs (64 x 64) staged in LDS with a K-pair-interleaved layout:
//   element (k, c) -> sW[(k/2)*PSTR + 2c + (k&1)]
// so each B fragment (rows k, k+1 @ column c) is ONE aligned b64 LDS
// load - no re-pairing movs. PSTR = 160 (== 32 mod 64) makes the two
// half-waves of a b64 read cover banks 0..31 / 32..63 -> conflict-free.
// Per K-step of 4: 1x global b64 (A) + 4x ds b64 (B) + 4x WMMA.
// M must be a multiple of 128 (always BL_ here), K a multiple of 4.
// =====================================================================
#define KSLAB 64
#define PSTR  160                     // words per K-pair row (128 data + pad)

__global__ __launch_bounds__(256)
void gemm_wmma_f32(const float* __restrict__ A, const float* __restrict__ W,
                   const float* __restrict__ bias, float* __restrict__ C,
                   int M, int N, int K, int flags)
{
  __shared__ float sW[(KSLAB/2) * PSTR];   // 5120 words = 20 KB

  const int tid  = threadIdx.x;
  const int wave = tid >> 5;           // 0..7 -> row sub-tile
  const int lane = tid & 31;
  const int half = lane >> 4;          // 0/1
  const int r    = lane & 15;

  const int m0 = (blockIdx.y << 7) + (wave << 4);   // this wave's 16 rows
  const int n0 = blockIdx.x << 6;                   // block's 64 cols

  v8f acc0 = {0.f,0.f,0.f,0.f,0.f,0.f,0.f,0.f};
  v8f acc1 = acc0, acc2 = acc0, acc3 = acc0;

  const float* __restrict__ arow = A + (size_t)(m0 + r) * K;

  for (int k0 = 0; k0 < K; k0 += KSLAB) {
    const int kslab = (K - k0 < KSLAB) ? (K - k0) : KSLAB;

    // ---- cooperative stage of W[k0:k0+kslab, n0:n0+64] into LDS ----
    // 4096 words / 256 threads = 4 x float4 per thread (global side)
    for (int base = tid * 4; base < KSLAB * 64; base += 1024) {
      const int kk = base >> 6;
      const int c  = base & 63;
      v4f w = {0.f, 0.f, 0.f, 0.f};
      if (kk < kslab && (n0 + c) < N) {
        // N is always a multiple of 4 here and c % 4 == 0 -> safe v4 load
        w = *(const v4f*)(W + (size_t)(k0 + kk) * N + n0 + c);
      }
      float* dst = &sW[(kk >> 1) * PSTR + (c << 1) + (kk & 1)];
      dst[0] = w.x;  dst[2] = w.y;  dst[4] = w.z;  dst[6] = w.w;
    }
    __syncthreads();

    // ---- guard-free WMMA inner loop ----
    #pragma unroll 2
    for (int kk = 0; kk < kslab; kk += 4) {
      const int ka = k0 + kk + (half << 1);
      const v2f a = *(const v2f*)(arow + ka);       // 8B aligned: ka even, K even
      // pair row index: (kk + 2*half)/2 ; fragment = one aligned b64 each
      const float* bsrc = &sW[((kk >> 1) + half) * PSTR + (r << 1)];
      const v2f b0 = *(const v2f*)(bsrc +  0);      // cols r
      const v2f b1 = *(const v2f*)(bsrc + 32);      // cols r+16
      const v2f b2 = *(const v2f*)(bsrc + 64);      // cols r+32
      const v2f b3 = *(const v2f*)(bsrc + 96);      // cols r+48
      acc0 = __builtin_amdgcn_wmma_f32_16x16x4_f32(false, a, false, b0, (short)0, acc0, false, false);
      acc1 = __builtin_amdgcn_wmma_f32_16x16x4_f32(false, a, false, b1, (short)0, acc1, false, false);
      acc2 = __builtin_amdgcn_wmma_f32_16x16x4_f32(false, a, false, b2, (short)0, acc2, false, false);
      acc3 = __builtin_amdgcn_wmma_f32_16x16x4_f32(false, a, false, b3, (short)0, acc3, false, false);
    }
    __syncthreads();
  }

  // ---- store epilogue: bias / residual-accumulate / column guards ----
  const bool doBias = (flags & GF_BIAS);
  const bool doAcc  = (flags & GF_ACC);
  #pragma unroll
  for (int t = 0; t < 4; ++t) {
    const v8f acc = (t == 0) ? acc0 : (t == 1) ? acc1 : (t == 2) ? acc2 : acc3;
    const int col = n0 + t * 16 + r;
    if (col >= N) continue;
    const float bv = doBias ? bias[col] : 0.f;
    #pragma unroll
    for (int i = 0; i < 8; ++i) {
      const int row = m0 + i + 8 * half;
      float v = acc[i] + bv;
      if (doAcc) v += C[(size_t)row * N + col];
      C[(size_t)row * N + col] = v;
    }
  }
}

// =====================================================================
// LayerNorm over DM_=128 columns; one wave32 per row, 4 elems/lane.
// =====================================================================
__global__ __launch_bounds__(256)
void layernorm_kernel(const float* __restrict__ x, const float* __restrict__ g,
                      const float* __restrict__ b, float* __restrict__ o, int nrows)
{
  const int wave = (int)((blockIdx.x * blockDim.x + threadIdx.x) >> 5);
  const int lane = threadIdx.x & 31;
  if (wave >= nrows) return;
  const float* __restrict__ row = x + (size_t)wave * DM_;
  float v[4];
  #pragma unroll
  for (int j = 0; j < 4; ++j) v[j] = row[lane + 32*j];
  float s = 0.f, sq = 0.f;
  #pragma unroll
  for (int j = 0; j < 4; ++j) { s += v[j]; sq += v[j]*v[j]; }
  #pragma unroll
  for (int off = 16; off > 0; off >>= 1) {
    s  += __shfl_xor(s,  off, 32);
    sq += __shfl_xor(sq, off, 32);
  }
  const float mean = s * (1.f/128.f);
  const float var  = sq * (1.f/128.f) - mean*mean;
  const float rstd = rsqrtf(var + 1e-5f);
  float* __restrict__ orow = o + (size_t)wave * DM_;
  #pragma unroll
  for (int j = 0; j < 4; ++j) {
    const int c = lane + 32*j;
    orow[c] = (v[j] - mean) * rstd * g[c] + b[c];
  }
}

// =====================================================================
// h = target*(1-m) + mask_token*m
// =====================================================================
__global__ __launch_bounds__(256)
void apply_mask_kernel(const float* __restrict__ target,
                       const unsigned char* __restrict__ mask,
                       const float* __restrict__ mask_token,
                       float* __restrict__ h)
{
  const int i = (int)(blockIdx.x * blockDim.x + threadIdx.x);
  if (i >= BL_ * DM_) return;
  const int row = i >> 7;
  const int c   = i & 127;
  const float m = mask[row] ? 1.f : 0.f;
  h[i] = target[i] * (1.f - m) + mask_token[c] * m;
}

__device__ __forceinline__ float silu_f(float x) {
  return x / (1.f + __expf(-x));
}

// =====================================================================
// Depthwise causal conv (K=4) + bias + SiLU. Reads xi half of xz (ld 512).
// =====================================================================
__global__ __launch_bounds__(256)
void conv_silu_kernel(const float* __restrict__ xz, const float* __restrict__ cw,
                      const float* __restrict__ cb, float* __restrict__ xic)
{
  const int i = (int)(blockIdx.x * blockDim.x + threadIdx.x);
  if (i >= BL_ * E_) return;
  const int e = i & (E_-1);
  const int t = (i >> 8) & (L_-1);
  const size_t base = (size_t)(i >> 8) * (2*E_) + e;   // row*512 + e (xi half)
  const float w0 = cw[e*K_+0], w1 = cw[e*K_+1], w2 = cw[e*K_+2], w3 = cw[e*K_+3];
  float acc = cb[e] + w3 * xz[base];
  if (t >= 1) acc += w2 * xz[base - 1*(2*E_)];
  if (t >= 2) acc += w1 * xz[base - 2*(2*E_)];
  if (t >= 3) acc += w0 * xz[base - 3*(2*E_)];
  xic[i] = silu_f(acc);
}

// =====================================================================
// delta = softplus(dt @ dt_w + dt_b);  dt = dbl[:, :8], K=8 -> scalar FMAs
// =====================================================================
__global__ __launch_bounds__(256)
void delta_kernel(const float* __restrict__ dbl, const float* __restrict__ dtw,
                  const float* __restrict__ dtb, float* __restrict__ delta)
{
  const int i = (int)(blockIdx.x * blockDim.x + threadIdx.x);
  if (i >= BL_ * E_) return;
  const int e   = i & (E_-1);
  const int row = i >> 8;
  const float* __restrict__ d = dbl + (size_t)row * XPN_;
  float acc = dtb[e];
  #pragma unroll
  for (int r = 0; r < R_; ++r) acc += d[r] * dtw[r*E_ + e];
  float sp;
  if (acc > 20.f)       sp = acc;
  else if (acc < -20.f) sp = __expf(acc);
  else                  sp = log1pf(__expf(acc));
  delta[i] = sp;
}

// =====================================================================
// Selective scan: grid = B_, block = E_ threads (one per channel).
// DS_=16 states in VGPRs; B/C rows staged in LDS per 64-step chunk;
// next chunk's operands prefetched (global_prefetch_b8) to hide HBM/L2
// latency behind the serial recurrence.
// =====================================================================
#define TCHUNK 64
__global__ __launch_bounds__(E_)
void scan_kernel(const float* __restrict__ delta, const float* __restrict__ xic,
                 const float* __restrict__ dbl, const float* __restrict__ Alog,
                 float* __restrict__ y)
{
  __shared__ float sB[TCHUNK][DS_];
  __shared__ float sC[TCHUNK][DS_];
  const int b = blockIdx.x;
  const int e = threadIdx.x;

  float a[DS_];
  #pragma unroll
  for (int n = 0; n < DS_; ++n) a[n] = -__expf(Alog[e*DS_ + n]);
  float hs[DS_];
  #pragma unroll
  for (int n = 0; n < DS_; ++n) hs[n] = 0.f;

  for (int t0 = 0; t0 < L_; t0 += TCHUNK) {
    // stage B (cols 8..23) and C (cols 24..39) of dbl for this chunk
    for (int j = threadIdx.x; j < TCHUNK * 2 * DS_; j += E_) {
      const int tt = j >> 5;
      const int c  = j & 31;
      const float v = dbl[((size_t)b*L_ + t0 + tt) * XPN_ + R_ + c];
      if (c < DS_) sB[tt][c] = v; else sC[tt][c - DS_] = v;
    }
    // prefetch next chunk while this one computes
    if (t0 + TCHUNK < L_) {
      const size_t nrow = (size_t)b*L_ + t0 + TCHUNK;
      #pragma unroll
      for (int tt = 0; tt < TCHUNK; tt += 16) {
        __builtin_prefetch(&delta[(nrow + tt) * E_ + e], 0, 1);
        __builtin_prefetch(&xic  [(nrow + tt) * E_ + e], 0, 1);
      }
      __builtin_prefetch(&dbl[(nrow + (e >> 2)) * XPN_ + R_ + ((e & 3) << 3)], 0, 1);
    }
    __syncthreads();

    for (int t = 0; t < TCHUNK; ++t) {
      const size_t row = (size_t)b*L_ + t0 + t;
      const float d  = delta[row*E_ + e];
      const float xv = xic[row*E_ + e];
      const float dx = d * xv;
      float acc = 0.f;
      #pragma unroll
      for (int n = 0; n < DS_; ++n) {
        const float dA = __expf(d * a[n]);
        hs[n] = dA * hs[n] + dx * sB[t][n];
        acc  += hs[n] * sC[t][n];
      }
      y[row*E_ + e] = acc;
    }
    __syncthreads();
  }
}

// =====================================================================
// y = (y + xic*D) * silu(z);  z is xz[:, 256:512]
// =====================================================================
__global__ __launch_bounds__(256)
void yfin_kernel(float* __restrict__ y, const float* __restrict__ xic,
                 const float* __restrict__ xz, const float* __restrict__ Dp)
{
  const int i = (int)(blockIdx.x * blockDim.x + threadIdx.x);
  if (i >= BL_ * E_) return;
  const int e   = i & (E_-1);
  const int row = i >> 8;
  const float zv = xz[(size_t)row * (2*E_) + E_ + e];
  y[i] = (y[i] + xic[i] * Dp[e]) * silu_f(zv);
}

// =====================================================================
// se = mean((pred-target)^2, axis=-1) * mask ; one wave32 per row
// =====================================================================
__global__ __launch_bounds__(256)
void se_kernel(const float* __restrict__ pred, const float* __restrict__ target,
               const unsigned char* __restrict__ mask, float* __restrict__ out, int nrows)
{
  const int wave = (int)((blockIdx.x * blockDim.x + threadIdx.x) >> 5);
  const int lane = threadIdx.x & 31;
  if (wave >= nrows) return;
  const float* __restrict__ p  = pred   + (size_t)wave * DM_;
  const float* __restrict__ tg = target + (size_t)wave * DM_;
  float s = 0.f;
  #pragma unroll
  for (int j = 0; j < 4; ++j) {
    const float d = p[lane + 32*j] - tg[lane + 32*j];
    s += d * d;
  }
  #pragma unroll
  for (int off = 16; off > 0; off >>= 1) s += __shfl_xor(s, off, 32);
  if (lane == 0) out[wave] = s * (1.f/128.f) * (mask[wave] ? 1.f : 0.f);
}

// =====================================================================
// Host-side orchestration
// =====================================================================
static inline dim3 gemm_grid(int M, int N) {
  return dim3((unsigned)((N + 63) / 64), (unsigned)(M / 128));
}

extern "C" void kernel_launch(void* const* d_in, const int* in_sizes, int n_in,
                              void* d_out, int out_size, void* d_ws, size_t ws_size,
                              hipStream_t stream)
{
  (void)in_sizes; (void)n_in; (void)out_size; (void)ws_size;

  const float*         x          = (const float*)d_in[0];
  const unsigned char* mask       = (const unsigned char*)d_in[1];   // bool array
  const float*         W_enc      = (const float*)d_in[2];
  const float*         b_enc      = (const float*)d_in[3];
  const float*         mask_token = (const float*)d_in[4];
  const float*         ln_g       = (const float*)d_in[5];
  const float*         ln_b       = (const float*)d_in[6];
  const float*         in_proj    = (const float*)d_in[7];
  const float*         conv_w     = (const float*)d_in[8];
  const float*         conv_b     = (const float*)d_in[9];
  const float*         x_proj     = (const float*)d_in[10];
  const float*         dt_w       = (const float*)d_in[11];
  const float*         dt_b       = (const float*)d_in[12];
  const float*         A_log      = (const float*)d_in[13];
  const float*         D_param    = (const float*)d_in[14];
  const float*         out_proj   = (const float*)d_in[15];
  const float*         fn_g       = (const float*)d_in[16];
  const float*         fn_b       = (const float*)d_in[17];
  const float*         head_w     = (const float*)d_in[18];
  const float*         head_b     = (const float*)d_in[19];
  float* out = (float*)d_out;

  // ---- workspace arena (floats) ----
  float* ws = (float*)d_ws;
  float* target = ws;                 ws += (size_t)BL_ * DM_;
  float* h      = ws;                 ws += (size_t)BL_ * DM_;
  float* hn     = ws;                 ws += (size_t)BL_ * DM_;
  float* xz     = ws;                 ws += (size_t)BL_ * (2*E_);   // [xi | z]
  float* xic    = ws;                 ws += (size_t)BL_ * E_;
  float* dblb   = ws;                 ws += (size_t)BL_ * XPN_;
  float* delta  = ws;                 ws += (size_t)BL_ * E_;
  float* ybuf   = ws;                 ws += (size_t)BL_ * E_;
  float* pred   = xz;                 // reuse xz region after layers

  const int nElemDM = BL_ * DM_;
  const int nElemE  = BL_ * E_;
  const dim3 blk256(256);
  const dim3 grdDM((nElemDM + 255) / 256);
  const dim3 grdE ((nElemE  + 255) / 256);
  const dim3 grdRowWave((BL_ * 32 + 255) / 256);   // one wave per row kernels

  // ---- encoder: target = x @ W_enc + b_enc ; h = masked mix ----
  gemm_wmma_f32<<<gemm_grid(BL_, DM_), blk256, 0, stream>>>(
      x, W_enc, b_enc, target, BL_, DM_, F_, GF_BIAS);
  apply_mask_kernel<<<grdDM, blk256, 0, stream>>>(target, mask, mask_token, h);

  // ---- Mamba layers ----
  for (int l = 0; l < NL_; ++l) {
    layernorm_kernel<<<grdRowWave, blk256, 0, stream>>>(
        h, ln_g + (size_t)l*DM_, ln_b + (size_t)l*DM_, hn, BL_);

    // xz = hn @ in_proj[l]          (BLx128 @ 128x512)
    gemm_wmma_f32<<<gemm_grid(BL_, 2*E_), blk256, 0, stream>>>(
        hn, in_proj + (size_t)l*DM_*(2*E_), nullptr, xz, BL_, 2*E_, DM_, 0);

    // depthwise causal conv + SiLU -> xic
    conv_silu_kernel<<<grdE, blk256, 0, stream>>>(
        xz, conv_w + (size_t)l*E_*K_, conv_b + (size_t)l*E_, xic);

    // dbl = xic @ x_proj[l]         (BLx256 @ 256x40)
    gemm_wmma_f32<<<gemm_grid(BL_, XPN_), blk256, 0, stream>>>(
        xic, x_proj + (size_t)l*E_*XPN_, nullptr, dblb, BL_, XPN_, E_, 0);

    // delta = softplus(dt @ dt_w + dt_b)
    delta_kernel<<<grdE, blk256, 0, stream>>>(
        dblb, dt_w + (size_t)l*R_*E_, dt_b + (size_t)l*E_, delta);

    // sequential selective scan -> ybuf
    scan_kernel<<<dim3(B_), dim3(E_), 0, stream>>>(
        delta, xic, dblb, A_log + (size_t)l*E_*DS_, ybuf);

    // ybuf = (ybuf + xic*D) * silu(z)
    yfin_kernel<<<grdE, blk256, 0, stream>>>(
        ybuf, xic, xz, D_param + (size_t)l*E_);

    // h += ybuf @ out_proj[l]       (BLx256 @ 256x128, residual accumulate)
    gemm_wmma_f32<<<gemm_grid(BL_, DM_), blk256, 0, stream>>>(
        ybuf, out_proj + (size_t)l*E_*DM_, nullptr, h, BL_, DM_, E_, GF_ACC);
  }

  // ---- final LN + head + masked MSE ----
  layernorm_kernel<<<grdRowWave, blk256, 0, stream>>>(h, fn_g, fn_b, hn, BL_);
  gemm_wmma_f32<<<gemm_grid(BL_, DM_), blk256, 0, stream>>>(
      hn, head_w, head_b, pred, BL_, DM_, DM_, GF_BIAS);
  se_kernel<<<grdRowWave, blk256, 0, stream>>>(pred, target, mask, out, BL_);
}